// MoEFusedTKG_22995254902978
// MI455X (gfx1250) — compile-verified
//
#include <hip/hip_runtime.h>

// MoE fused TKG decode for MI455X (gfx1250, wave32).
// Memory-bound (~1.7 GB weights / call); GEMMs via V_WMMA_F32_16X16X4_F32.
// A-chunk staging double-buffered via CDNA5 TDM (tensor_load_to_lds) and async-to-LDS loads.

typedef __attribute__((ext_vector_type(2))) float v2f;
typedef __attribute__((ext_vector_type(4))) float v4f;
typedef __attribute__((ext_vector_type(8))) float v8f;
typedef __attribute__((ext_vector_type(4))) unsigned int su4;
typedef __attribute__((ext_vector_type(8))) unsigned int su8;

#define HDIM 2048
#define NEXP 16
#define IDIM 4096
#define NTOK 32
#define KC   128
#define LSTR (KC + 4)   // LDS row stride (floats); 132*4B = 528B, 16B aligned, bank-conflict free
#define BUFB (NTOK * LSTR * 4)  // bytes per LDS staging buffer

// ---- TDM: load a 32-row x 128-col f32 tile (row stride = strideElems) into LDS,
// ---- inserting 4 DWORDs of padding after every 128 DWORDs (=> LDS stride 132).
__device__ __forceinline__ void tdm_load_tile_32x128(const float* gptr, unsigned lds_byte_addr,
                                                     unsigned strideElems) {
    unsigned long long ga = (unsigned long long)(uintptr_t)gptr;
    su4 g0;
    g0[0] = 1u;                                              // count=1 (valid), user mode
    g0[1] = lds_byte_addr;                                   // lds_addr
    g0[2] = (unsigned)(ga & 0xFFFFFFFFu);                    // global_addr[31:0]
    g0[3] = ((unsigned)(ga >> 32) & 0x01FFFFFFu) | (2u << 30); // global_addr[56:32] | type=2
    su8 g1;
    g1[0] = (2u << 16)      // data_size = 4B
          | (1u << 20)      // pad_enable
          | (6u << 22)      // pad_interval: 128 DWORDs
          | (3u << 25);     // pad_amount: 4 DWORDs
    g1[1] = (strideElems & 0xFFFFu) << 16;                   // tensor_dim0[15:0]
    g1[2] = (strideElems >> 16) | (32u << 16);               // tensor_dim0[31:16] | tensor_dim1=32
    g1[3] = (128u << 16);                                    // tile_dim0 = 128
    g1[4] = 32u;                                             // tile_dim1 = 32, tile_dim2 = 0
    g1[5] = strideElems;                                     // tensor_dim0_stride[31:0]
    g1[6] = 0u;
    g1[7] = 0u;
    asm volatile("tensor_load_to_lds %0, %1" :: "s"(g0), "s"(g1) : "memory");
}

// ---- Async global->LDS 16B copy (per-lane), tracked with ASYNCcnt.
__device__ __forceinline__ void async_load_b128(unsigned lds_byte_addr, const float* gaddr) {
    asm volatile("global_load_async_to_lds_b128 %0, %1, off"
                 :: "v"(lds_byte_addr), "v"(gaddr) : "memory");
}

// ---------------- Kernel A: RMSNorm + router softmax/top-4 ----------------
__global__ __launch_bounds__(256) void moe_norm_router_k(
    const float* __restrict__ hs, const float* __restrict__ lnw,
    const float* __restrict__ rw, float* __restrict__ xn, float* __restrict__ aff)
{
    __shared__ float lx[HDIM];
    __shared__ float red[256];
    __shared__ float lg[NEXP];
    const int t = blockIdx.x, tid = threadIdx.x;

    float s = 0.f;
    #pragma unroll
    for (int j = 0; j < HDIM / 256; ++j) {
        float v = hs[(size_t)t * HDIM + tid + j * 256];
        lx[tid + j * 256] = v;
        s += v * v;
    }
    red[tid] = s; __syncthreads();
    for (int off = 128; off > 0; off >>= 1) {
        if (tid < off) red[tid] += red[tid + off];
        __syncthreads();
    }
    const float scale = rsqrtf(red[0] / (float)HDIM + 1e-6f);

    #pragma unroll
    for (int j = 0; j < HDIM / 256; ++j) {
        int c = tid + j * 256;
        float v = lx[c] * scale * lnw[c];
        lx[c] = v;
        xn[(size_t)t * HDIM + c] = v;
    }
    __syncthreads();

    const int e = tid >> 4, k = tid & 15;
    float p = 0.f;
    for (int m = 0; m < HDIM / 16; ++m) {
        int row = k + m * 16;
        p += lx[row] * rw[(size_t)row * NEXP + e];
    }
    red[e * 16 + k] = p; __syncthreads();
    if (tid < NEXP) {
        float sum = 0.f;
        #pragma unroll
        for (int i = 0; i < 16; ++i) sum += red[tid * 16 + i];
        lg[tid] = sum;
    }
    __syncthreads();
    if (tid == 0) {
        float mx = lg[0];
        for (int i = 1; i < NEXP; ++i) mx = fmaxf(mx, lg[i]);
        float pr[NEXP], a[NEXP], ssum = 0.f;
        for (int i = 0; i < NEXP; ++i) { pr[i] = __expf(lg[i] - mx); ssum += pr[i]; }
        float inv = 1.f / ssum;
        for (int i = 0; i < NEXP; ++i) { pr[i] *= inv; a[i] = 0.f; }
        for (int kk = 0; kk < 4; ++kk) {
            int bi = 0; float bv = pr[0];
            for (int i = 1; i < NEXP; ++i) if (pr[i] > bv) { bv = pr[i]; bi = i; }
            a[bi] = bv; pr[bi] = -1.f;
        }
        for (int i = 0; i < NEXP; ++i) aff[t * NEXP + i] = a[i];
    }
}

// -------- Kernel B: per-expert (and shared) gate/up GEMM + SwiGLU, affinity folded in --------
// grid.x = 17 * (IDIM/64); 8 waves/block; A staged through double-buffered LDS via TDM.
__global__ __launch_bounds__(256) void moe_gateup_k(
    const float* __restrict__ xn, const float* __restrict__ aff,
    const float* __restrict__ gup, const float* __restrict__ sgw,
    const float* __restrict__ suw, float* __restrict__ h, float* __restrict__ sh)
{
    __shared__ float lA[2][NTOK * LSTR];
    const int bx = blockIdx.x;
    const int e = bx >> 6;          // expert (16 == shared)
    const int tile = bx & 63;
    const int tid = threadIdx.x;
    const int lane = tid & 31, wid = tid >> 5;
    const int m = wid & 1, n = wid >> 1;
    const int laneIdx = lane & 15, laneHalf = lane >> 4;
    const int colBase = tile * 64;
    const int colJ = n * 16 + laneIdx;

    const float* Wg; const float* Wu; size_t ldw; float* hout;
    if (e < NEXP) {
        const float* base = gup + (size_t)e * HDIM * (2 * IDIM);
        Wg = base + colBase;
        Wu = base + IDIM + colBase;
        ldw = 2 * IDIM;
        hout = h + (size_t)e * NTOK * IDIM;
    } else {
        Wg = sgw + colBase; Wu = suw + colBase; ldw = IDIM; hout = sh;
    }

    const unsigned ldsBase = (unsigned)(uintptr_t)(void*)&lA[0][0]; // LDS byte address = low 32 bits of the pointer
    v8f cg = {}; v8f cu = {};
    const int aRow = (m * 16 + laneIdx) * LSTR;

    if (wid == 0) tdm_load_tile_32x128(xn, ldsBase, HDIM);  // prologue: chunk 0 -> buf 0
    int buf = 0;

    for (int kc = 0; kc < HDIM; kc += KC) {
        if (wid == 0) {
            if (kc + KC < HDIM) {
                tdm_load_tile_32x128(xn + kc + KC, ldsBase + (unsigned)(buf ^ 1) * BUFB, HDIM);
                __builtin_amdgcn_s_wait_tensorcnt(1);   // chunk kc landed; kc+KC in flight
            } else {
                __builtin_amdgcn_s_wait_tensorcnt(0);
            }
        }
        __syncthreads();                                // data ready + prev readers done
        const float* lAc = &lA[buf][0];
        const float* wg = Wg + (size_t)kc * ldw;
        const float* wu = Wu + (size_t)kc * ldw;
        #pragma unroll 4
        for (int k4 = 0; k4 < KC; k4 += 4) {
            const int krow = k4 + 2 * laneHalf;
            v2f a;  a.x  = lAc[aRow + krow];              a.y  = lAc[aRow + krow + 1];
            v2f bg; bg.x = wg[(size_t)krow * ldw + colJ]; bg.y = wg[(size_t)(krow + 1) * ldw + colJ];
            v2f bu; bu.x = wu[(size_t)krow * ldw + colJ]; bu.y = wu[(size_t)(krow + 1) * ldw + colJ];
            cg = __builtin_amdgcn_wmma_f32_16x16x4_f32(false, a, false, bg, (short)0, cg, false, false);
            cu = __builtin_amdgcn_wmma_f32_16x16x4_f32(false, a, false, bu, (short)0, cu, false, false);
        }
        __syncthreads();                                // all done reading buf before overwrite
        buf ^= 1;
    }

    const int iCol = colBase + colJ;
    #pragma unroll
    for (int j = 0; j < 8; ++j) {
        int trow = m * 16 + j + 8 * laneHalf;
        float af = (e < NEXP) ? aff[trow * NEXP + e] : 1.f;
        float g = cg[j], u = cu[j];
        float hv = g * __builtin_amdgcn_rcpf(1.f + __expf(-g)) * u * af;
        hout[(size_t)trow * IDIM + iCol] = hv;
    }
}

// -------- Kernel C: down-projection partials; A staged via double-buffered async-to-LDS --------
// grid.x = 17 * (HDIM/64)
__global__ __launch_bounds__(256) void moe_down_k(
    const float* __restrict__ h, const float* __restrict__ sh,
    const float* __restrict__ dwn, const float* __restrict__ sdw,
    float* __restrict__ part)
{
    __shared__ float lA[2][NTOK * LSTR];
    const int bx = blockIdx.x;
    const int e = bx >> 5;
    const int tile = bx & 31;
    const int tid = threadIdx.x;
    const int lane = tid & 31, wid = tid >> 5;
    const int m = wid & 1, n = wid >> 1;
    const int laneIdx = lane & 15, laneHalf = lane >> 4;
    const int colBase = tile * 64;
    const int colJ = n * 16 + laneIdx;

    const float* A; const float* W;
    if (e < NEXP) { A = h + (size_t)e * NTOK * IDIM; W = dwn + (size_t)e * IDIM * HDIM; }
    else          { A = sh;                          W = sdw; }

    const unsigned ldsBase = (unsigned)(uintptr_t)(void*)&lA[0][0];
    v8f c = {};
    const int aRow = (m * 16 + laneIdx) * LSTR;
    const int row0 = tid >> 5, c40 = tid & 31;   // per-thread staging coords (i-step adds rows)

    // prologue: chunk 0 -> buf 0 (4 x 16B per thread)
    #pragma unroll
    for (int i = 0; i < 4; ++i) {
        int row = row0 + i * 8;
        async_load_b128(ldsBase + (unsigned)(row * LSTR + c40 * 4) * 4u,
                        A + (size_t)row * IDIM + c40 * 4);
    }
    int buf = 0;

    for (int kc = 0; kc < IDIM; kc += KC) {
        if (kc + KC < IDIM) {
            unsigned dstBase = ldsBase + (unsigned)(buf ^ 1) * BUFB;
            #pragma unroll
            for (int i = 0; i < 4; ++i) {
                int row = row0 + i * 8;
                async_load_b128(dstBase + (unsigned)(row * LSTR + c40 * 4) * 4u,
                                A + (size_t)row * IDIM + kc + KC + c40 * 4);
            }
            asm volatile("s_wait_asynccnt 0x4" ::: "memory");  // chunk kc done; kc+KC in flight
        } else {
            asm volatile("s_wait_asynccnt 0x0" ::: "memory");
        }
        __syncthreads();
        const float* lAc = &lA[buf][0];
        const float* w = W + (size_t)kc * HDIM + colBase;
        #pragma unroll 4
        for (int k4 = 0; k4 < KC; k4 += 4) {
            const int krow = k4 + 2 * laneHalf;
            v2f a; a.x = lAc[aRow + krow];              a.y = lAc[aRow + krow + 1];
            v2f b; b.x = w[(size_t)krow * HDIM + colJ]; b.y = w[(size_t)(krow + 1) * HDIM + colJ];
            c = __builtin_amdgcn_wmma_f32_16x16x4_f32(false, a, false, b, (short)0, c, false, false);
        }
        __syncthreads();
        buf ^= 1;
    }

    #pragma unroll
    for (int j = 0; j < 8; ++j) {
        int trow = m * 16 + j + 8 * laneHalf;
        part[((size_t)e * NTOK + trow) * HDIM + colBase + colJ] = c[j];
    }
}

// -------- Kernel D: deterministic reduction of 17 partial slabs --------
__global__ __launch_bounds__(256) void moe_reduce_k(const float* __restrict__ part,
                                                   float* __restrict__ out)
{
    const int i = blockIdx.x * 256 + threadIdx.x;   // 0 .. NTOK*HDIM-1
    float s = 0.f;
    #pragma unroll
    for (int e = 0; e < NEXP + 1; ++e) s += part[(size_t)e * NTOK * HDIM + i];
    out[i] = s;
}

extern "C" void kernel_launch(void* const* d_in, const int* in_sizes, int n_in,
                              void* d_out, int out_size, void* d_ws, size_t ws_size,
                              hipStream_t stream) {
    const float* hs  = (const float*)d_in[0];
    const float* lnw = (const float*)d_in[1];
    const float* rw  = (const float*)d_in[2];
    const float* gup = (const float*)d_in[3];
    const float* dwn = (const float*)d_in[4];
    const float* sgw = (const float*)d_in[5];
    const float* suw = (const float*)d_in[6];
    const float* sdw = (const float*)d_in[7];
    float* out = (float*)d_out;

    float* ws   = (float*)d_ws;
    float* xn   = ws;                                  // 32*2048
    float* aff  = xn  + (size_t)NTOK * HDIM;           // 32*16
    float* h    = aff + (size_t)NTOK * NEXP;           // 16*32*4096 (affinity pre-scaled)
    float* sh   = h   + (size_t)NEXP * NTOK * IDIM;    // 32*4096
    float* part = sh  + (size_t)NTOK * IDIM;           // 17*32*2048

    moe_norm_router_k<<<NTOK, 256, 0, stream>>>(hs, lnw, rw, xn, aff);
    moe_gateup_k<<<(NEXP + 1) * (IDIM / 64), 256, 0, stream>>>(xn, aff, gup, sgw, suw, h, sh);
    moe_down_k<<<(NEXP + 1) * (HDIM / 64), 256, 0, stream>>>(h, sh, dwn, sdw, part);
    moe_reduce_k<<<(NTOK * HDIM) / 256, 256, 0, stream>>>(part, out);
}